// BilinearSamplingLayer_77773267796584
// MI455X (gfx1250) — compile-verified
//
#include <hip/hip_runtime.h>

// Dense image warp (bilinear sampling), NHWC f32.
//   image: [B,256,256,32] f32, flow: [B,256,256,2] f32, out: [B,256,256,32] f32
//   out[b,i,j,c] = bilerp(image[b], y = i - 256*flow[b,i,j,0], x = j - 256*flow[b,i,j,1])
//   TF clamping: floor clipped to [0, 254], alpha clipped to [0,1].
//
// Memory-bound kernel (~528 MiB traffic -> ~23 us floor @ 23.3 TB/s).
// 8 lanes per pixel, one float4 channel chunk per lane (C=32 -> 8 chunks).

typedef __attribute__((ext_vector_type(2))) float f32x2;
typedef __attribute__((ext_vector_type(4))) float f32x4;

#define BDIM 256

__global__ __launch_bounds__(BDIM) void
BilinearSamplingLayer_77773267796584_kernel(const float* __restrict__ image,
                                            const float* __restrict__ flow,
                                            float* __restrict__ out,
                                            int npix)  // B*256*256
{
    const int gid   = blockIdx.x * BDIM + threadIdx.x;
    const int pixel = gid >> 3;   // 8 float4 chunks per pixel
    const int chunk = gid & 7;
    if (pixel >= npix) return;

    const int b = pixel >> 16;          // / (256*256)
    const int i = (pixel >> 8) & 255;
    const int j = pixel & 255;

    // Flow is stream-once: non-temporal load (8 lanes of a pixel-group dedupe
    // to a single coalesced request within the wave).
    const f32x2 f = __builtin_nontemporal_load(
        reinterpret_cast<const f32x2*>(flow) + pixel);

    // Speculatively prefetch the identity-location image line while the
    // dependent gather address is still being computed; displacements are a
    // few pixels, so this line is usually the (or adjacent to the) target.
    // Lowers to global_prefetch_b8 on gfx1250 (no counter, no VGPR return).
    const int ident = ((pixel << 5) + (chunk << 2));  // pixel*32 + chunk*4
    __builtin_prefetch(image + ident, 0, 3);

    // Scaled flow (IMAGE_SIZE = 256), query coords.
    const float qy = (float)i - f.x * 256.0f;
    const float qx = (float)j - f.y * 256.0f;

    // TF-style clamping: floor in [0, H-2], alpha in [0, 1].
    const float fy = fminf(fmaxf(floorf(qy), 0.0f), 254.0f);
    const float fx = fminf(fmaxf(floorf(qx), 0.0f), 254.0f);
    const float ay = fminf(fmaxf(qy - fy, 0.0f), 1.0f);
    const float ax = fminf(fmaxf(qx - fx, 0.0f), 1.0f);
    const int iy = (int)fy;
    const int ix = (int)fx;

    // Gather base in floats: ((b*256 + iy)*256 + ix)*32 + chunk*4
    const int base = (((b << 8) + iy) << 13) + (ix << 5) + (chunk << 2);
    const float* __restrict__ p = image + base;

    // Four 16B taps -> global_load_b128; heavy spatial reuse, keep cached (RT).
    const f32x4 tl = *reinterpret_cast<const f32x4*>(p);
    const f32x4 tr = *reinterpret_cast<const f32x4*>(p + 32);     // ix+1
    const f32x4 bl = *reinterpret_cast<const f32x4*>(p + 8192);   // iy+1
    const f32x4 br = *reinterpret_cast<const f32x4*>(p + 8224);   // iy+1, ix+1

    const f32x4 top = tl + ax * (tr - tl);
    const f32x4 bot = bl + ax * (br - bl);
    const f32x4 res = top + ay * (bot - top);

    // Output is write-once: non-temporal b128 store (no write-allocate
    // pollution of the caches serving the gather taps).
    __builtin_nontemporal_store(res,
        reinterpret_cast<f32x4*>(out) + gid);
}

extern "C" void kernel_launch(void* const* d_in, const int* in_sizes, int n_in,
                              void* d_out, int out_size, void* d_ws, size_t ws_size,
                              hipStream_t stream) {
    (void)n_in; (void)d_ws; (void)ws_size; (void)out_size;

    const float* image = (const float*)d_in[0];
    const float* flow  = (const float*)d_in[1];
    float*       out   = (float*)d_out;

    // B derived from the image's flat size; H=W=256, C=32 fixed by reference.
    const int B    = in_sizes[0] / (256 * 256 * 32);
    const int npix = B * 256 * 256;
    const int nthreads = npix * 8;                // 8 float4 chunks per pixel
    const int nblocks  = (nthreads + BDIM - 1) / BDIM;

    BilinearSamplingLayer_77773267796584_kernel<<<nblocks, BDIM, 0, stream>>>(
        image, flow, out, npix);
}